// DRConv_26147760898633
// MI455X (gfx1250) — compile-verified
//
#include <hip/hip_runtime.h>
#include <math.h>

// Problem constants (match reference)
#define IN_C   128
#define OUT_C  128
#define GROUPS 8
#define NUMW   8
#define HDIM   56
#define WDIM   56
#define BDIM   8
#define KDIM   (IN_C * 9)        // 1152
#define PIX    (HDIM * WDIM)     // 3136

// GEMM tiling: one slice = 4 input channels x 9 taps = 36 k-values
#define KB     36
#define TN     64                // pixels per block tile
#define ALST   36                // A LDS row stride (144B, 16B-aligned, conflict-free)
#define BLST   38                // B LDS row stride (152B, 8B-aligned, conflict-free stores)
#define NSLICE 32                // KDIM / KB
#define TOTS   (GROUPS * NSLICE) // 256 slices per block
#define AV4    (128 * ALST / 4)  // 1152 v4f elements in A tile

typedef __attribute__((ext_vector_type(2))) float v2f;
typedef __attribute__((ext_vector_type(4))) float v4f;
typedef __attribute__((ext_vector_type(8))) float v8f;

// ---------------------------------------------------------------------------
// 1) Global average pool: mean[b*128+c] = mean over 56*56
// ---------------------------------------------------------------------------
__global__ void k_gap(const float* __restrict__ in, float* __restrict__ mean) {
    int bc = blockIdx.x;                       // 0..1023
    const float* p = in + (size_t)bc * PIX;
    float s = 0.f;
    for (int i = threadIdx.x; i < PIX; i += 256) s += p[i];
    __shared__ float red[256];
    red[threadIdx.x] = s;
    __syncthreads();
    for (int st = 128; st > 0; st >>= 1) {
        if (threadIdx.x < st) red[threadIdx.x] += red[threadIdx.x + st];
        __syncthreads();
    }
    if (threadIdx.x == 0) mean[bc] = red[0] * (1.0f / PIX);
}

// ---------------------------------------------------------------------------
// 2) Routing FC: coef[b,t] = 2*sigmoid(mean[b]·rw[t] + rb[t]) / NUMW  (=0.25*sig)
// ---------------------------------------------------------------------------
__global__ void k_route(const float* __restrict__ mean,
                        const float* __restrict__ rw,
                        const float* __restrict__ rb,
                        float* __restrict__ coef) {
    int b = blockIdx.x, t = threadIdx.x;       // t < 64
    const float* m = mean + b * IN_C;
    const float* w = rw + t * IN_C;
    float z = rb[t];
    for (int c = 0; c < IN_C; ++c) z += m[c] * w[c];
    coef[b * 64 + t] = 0.25f / (1.0f + __expf(-z));
}

// ---------------------------------------------------------------------------
// 3) Blend templates: wbuf[bg][ok] = sum_j coef[bg][j] * tmpl[ok][j]
// ---------------------------------------------------------------------------
__global__ void k_blend(const float* __restrict__ tmpl,
                        const float* __restrict__ coef,
                        float* __restrict__ wbuf) {
    __shared__ float cl[512];
    cl[threadIdx.x]       = coef[threadIdx.x];
    cl[threadIdx.x + 256] = coef[threadIdx.x + 256];
    __syncthreads();
    int ok = blockIdx.x * 256 + threadIdx.x;   // < 147456
    v4f t0 = *(const v4f*)(tmpl + (size_t)ok * 8);
    v4f t1 = *(const v4f*)(tmpl + (size_t)ok * 8 + 4);
    for (int bg = 0; bg < 64; ++bg) {
        const float* cc = &cl[bg * 8];
        float a = t0.x*cc[0] + t0.y*cc[1] + t0.z*cc[2] + t0.w*cc[3]
                + t1.x*cc[4] + t1.y*cc[5] + t1.z*cc[6] + t1.w*cc[7];
        wbuf[(size_t)bg * (OUT_C * KDIM) + ok] = a;
    }
}

// ---------------------------------------------------------------------------
// 4) Per-pixel routing probs: softmax over 8 groups (use_alpha) or one-hot(mask)
// ---------------------------------------------------------------------------
__global__ void k_probs(const float* __restrict__ Alpha,
                        const int* __restrict__ mask,
                        const int* __restrict__ use_alpha,
                        float* __restrict__ probs) {
    int idx = blockIdx.x * 256 + threadIdx.x;  // b*PIX + p
    if (idx >= BDIM * PIX) return;
    int b = idx / PIX, p = idx - b * PIX;
    float* o = probs + (size_t)b * GROUPS * PIX + p;
    if (*use_alpha) {
        const float* a = Alpha + (size_t)b * GROUPS * PIX + p;
        float v[8], mx = -1e30f;
        for (int g = 0; g < 8; ++g) { v[g] = a[g * PIX]; mx = fmaxf(mx, v[g]); }
        float s = 0.f;
        for (int g = 0; g < 8; ++g) { v[g] = __expf(v[g] - mx); s += v[g]; }
        float inv = 1.0f / s;
        for (int g = 0; g < 8; ++g) o[g * PIX] = v[g] * inv;
    } else {
        int mg = mask[idx];
        for (int g = 0; g < 8; ++g) o[g * PIX] = (g == mg) ? 1.0f : 0.0f;
    }
}

// ---------------------------------------------------------------------------
// 5) Main GEMM, software-pipelined, division-free im2col:
//    slice = 4 channels x 9 taps; each thread owns one channel's 9 taps for
//    one pixel -> taps/validity precomputed once, gather = 9 predicated loads.
// ---------------------------------------------------------------------------
__global__ __launch_bounds__(256, 1)
void k_drconv(const float* __restrict__ in,     // [B,128,56,56]
              const float* __restrict__ wbuf,   // [64][128][1152]
              const float* __restrict__ probs,  // [B][8][3136]
              const float* __restrict__ bias,   // [128]
              float* __restrict__ out) {        // [B][128][3136]
    __shared__ __align__(16) float Alds[128 * ALST];
    __shared__ __align__(16) float Blds[TN * BLST];

    const int tid  = threadIdx.x;
    const int wave = tid >> 5;
    const int lane = tid & 31;
    const int lm   = lane & 15;      // M (A) / N (B,C) lane index
    const int lh   = lane >> 4;      // K-half (A,B) / M-half (C)
    const int b    = blockIdx.y;
    const int pix0 = blockIdx.x * TN;
    const float* inb = in + (size_t)b * IN_C * PIX;

    // ---- A staging: flat v4f index over 128x36 tile (9 v4f per row) -------
    int agoff[5], aloff[5];          // float offsets (global w/o k0; LDS)
#pragma unroll
    for (int j = 0; j < 5; ++j) {
        int f   = tid + 256 * j;     // v4f index, valid if f < 1152
        int row = f / 9, c4 = f - row * 9;
        agoff[j] = row * KDIM + c4 * 4;
        aloff[j] = row * ALST + c4 * 4;
    }
    const bool a5 = (tid < AV4 - 1024);          // 5th element for tid < 128

    // ---- B staging: fixed pixel, fixed 9-tap window, channel varies -------
    const int q    = tid >> 6;       // channel sub-index within slice (0..3)
    const int bpix = tid & 63;
    const int pix  = pix0 + bpix;
    const int h    = pix / WDIM, w = pix - h * WDIM;
    const int center = h * WDIM + w;
    bool bval[9];
    int  boff[9];
#pragma unroll
    for (int i = 0; i < 9; ++i) {
        const int kh = i / 3, kw = i % 3;        // compile-time after unroll
        boff[i] = (kh - 1) * WDIM + (kw - 1);
        bval[i] = (h + kh - 1 >= 0) && (h + kh - 1 < HDIM) &&
                  (w + kw - 1 >= 0) && (w + kw - 1 < WDIM);
    }

    v8f oacc[4] = {};
    v8f cacc[4] = {};
    v4f areg[5];
    float breg[9];

    auto load_slice = [&](int s) {
        int g  = s >> 5;             // NSLICE = 32
        int sg = s & 31;
        const float* Ws = wbuf + (size_t)(b * GROUPS + g) * (OUT_C * KDIM)
                        + sg * KB;   // k0 = 36*sg (16B-aligned: 144*sg bytes)
#pragma unroll
        for (int j = 0; j < 4; ++j) areg[j] = *(const v4f*)(Ws + agoff[j]);
        if (a5) areg[4] = *(const v4f*)(Ws + agoff[4]);

        const float* p = inb + (size_t)(4 * sg + q) * PIX + center;
#pragma unroll
        for (int i = 0; i < 9; ++i)
            breg[i] = bval[i] ? p[boff[i]] : 0.f;
    };

    auto store_slice = [&]() {
#pragma unroll
        for (int j = 0; j < 4; ++j) *(v4f*)(&Alds[aloff[j]]) = areg[j];
        if (a5) *(v4f*)(&Alds[aloff[4]]) = areg[4];
        float* bd = &Blds[bpix * BLST + q * 9];
#pragma unroll
        for (int i = 0; i < 9; ++i) bd[i] = breg[i];
    };

    load_slice(0);
    for (int s = 0; s < TOTS; ++s) {
        __syncthreads();                 // LDS consumers of slice s-1 done
        store_slice();                   // regs(s) -> LDS
        if (s + 1 < TOTS) load_slice(s + 1);   // issue next slice's loads
        __syncthreads();                 // LDS tile (s) visible

        // WMMA: wave owns M rows [16*wave, 16*wave+16) x 64 px; 36 k per stage
#pragma unroll
        for (int kk = 0; kk < KB; kk += 4) {
            // A 16x4: lane lm = M row, lh selects k-pair {0,1}/{2,3}
            v2f av = *(const v2f*)&Alds[(wave * 16 + lm) * ALST + kk + 2 * lh];
#pragma unroll
            for (int t = 0; t < 4; ++t) {
                // B 4x16: lane lm = N col, lh selects k-pair
                v2f bv = *(const v2f*)&Blds[(t * 16 + lm) * BLST + kk + 2 * lh];
                cacc[t] = __builtin_amdgcn_wmma_f32_16x16x4_f32(
                    false, av, false, bv, (short)0, cacc[t], false, false);
            }
        }

        // End of group: fold probs-weighted result, reset group accumulator
        if (((s + 1) & (NSLICE - 1)) == 0) {
            int g = s >> 5;
            const float* pr = probs + (size_t)(b * GROUPS + g) * PIX + pix0;
#pragma unroll
            for (int t = 0; t < 4; ++t) {
                float p = pr[t * 16 + lm];
#pragma unroll
                for (int v = 0; v < 8; ++v) {
                    oacc[t][v] += p * cacc[t][v];
                    cacc[t][v] = 0.f;
                }
            }
        }
    }

    // Store: C/D layout -> o = 16*wave + 8*lh + v, pixel = pix0 + 16t + lm
#pragma unroll
    for (int t = 0; t < 4; ++t) {
#pragma unroll
        for (int v = 0; v < 8; ++v) {
            int o = wave * 16 + 8 * lh + v;
            out[((size_t)b * OUT_C + o) * PIX + pix0 + t * 16 + lm] =
                oacc[t][v] + bias[o];
        }
    }
}

// ---------------------------------------------------------------------------
extern "C" void kernel_launch(void* const* d_in, const int* in_sizes, int n_in,
                              void* d_out, int out_size, void* d_ws, size_t ws_size,
                              hipStream_t stream) {
    const float* inputs    = (const float*)d_in[0];
    const int*   mask      = (const int*)  d_in[1];
    const float* Alpha     = (const float*)d_in[2];
    const float* tmpl      = (const float*)d_in[3];
    const float* rw        = (const float*)d_in[4];
    const float* rb        = (const float*)d_in[5];
    const float* bias      = (const float*)d_in[6];
    const int*   use_alpha = (const int*)  d_in[7];
    float* out = (float*)d_out;

    // Workspace layout (~38.6 MB total)
    char*  ws    = (char*)d_ws;
    float* wbuf  = (float*)ws;                                    // 64*147456 f32
    float* probs = (float*)(ws + (size_t)64 * OUT_C * KDIM * 4);  // 200704 f32
    float* mean  = probs + (size_t)BDIM * GROUPS * PIX;           // 1024 f32
    float* coef  = mean + BDIM * IN_C;                            // 512 f32

    k_gap   <<<dim3(BDIM * IN_C), dim3(256), 0, stream>>>(inputs, mean);
    k_route <<<dim3(BDIM),        dim3(64),  0, stream>>>(mean, rw, rb, coef);
    k_blend <<<dim3((OUT_C * KDIM) / 256), dim3(256), 0, stream>>>(tmpl, coef, wbuf);
    k_probs <<<dim3((BDIM * PIX + 255) / 256), dim3(256), 0, stream>>>(Alpha, mask, use_alpha, probs);
    k_drconv<<<dim3(PIX / TN, BDIM), dim3(256), 0, stream>>>(inputs, wbuf, probs, bias, out);
}